// DynamicGraphConstructor_18433999634862
// MI455X (gfx1250) — compile-verified
//
#include <hip/hip_runtime.h>
#include <hip/hip_bf16.h>
#include <stdint.h>

// Problem constants (fixed by the reference setup)
constexpr int B = 16;
constexpr int N = 8192;
constexpr int C = 256;
constexpr int K = 4096;           // N * K_RATIO

// Output layout (flat floats, tuple return order):
//   [0]               selected_features  B*K*C
//   [OFF_IDX]         topk_indices       B*K   (int32 bit patterns)
//   [OFF_ADJ]         selected_adjacency B*K*K
//   [OFF_SC]          topk_scores        B*K
constexpr size_t OFF_FEAT = 0;
constexpr size_t OFF_IDX  = (size_t)B * K * C;
constexpr size_t OFF_ADJ  = OFF_IDX + (size_t)B * K;
constexpr size_t OFF_SC   = OFF_ADJ + (size_t)B * K * K;

// Native clang vector type (required by __builtin_nontemporal_*)
typedef float v4f __attribute__((ext_vector_type(4)));

// ---------------------------------------------------------------------------
// CDNA5 async global->LDS copy path (ASYNCcnt-tracked).
// Signature confirmed by the round-1 diagnostic:
//   void __builtin_amdgcn_global_load_async_to_lds_b128(
//       v4i addrspace(1)* gsrc, v4i addrspace(3)* ldst, imm offset, imm cpol)
// ---------------------------------------------------------------------------
typedef int v4i __attribute__((vector_size(16)));
typedef __attribute__((address_space(1))) v4i* gv4i_t;   // global
typedef __attribute__((address_space(3))) v4i* lv4i_t;   // LDS

#if defined(__gfx1250__) && __has_builtin(__builtin_amdgcn_global_load_async_to_lds_b128)
#define HAVE_ASYNC_LDS 1
#else
#define HAVE_ASYNC_LDS 0
#endif

#if defined(__gfx1250__) && __has_builtin(__builtin_amdgcn_s_wait_asynccnt)
#define WAIT_ASYNC(n) __builtin_amdgcn_s_wait_asynccnt(n)
#elif defined(__gfx1250__)
#define WAIT_ASYNC(n) asm volatile("s_wait_asynccnt " #n ::: "memory")
#else
#define WAIT_ASYNC(n)
#endif

// ---------------------------------------------------------------------------
// Kernel 1: per-batch bitonic sort of (score, index) pairs in LDS.
// One workgroup per batch. 8192 pairs = 64 KB LDS (well under 320 KB/WGP).
// Produces descending scores with original indices == jax.lax.top_k order.
// ---------------------------------------------------------------------------
__global__ __launch_bounds__(1024)
void topk_sort_kernel(const float* __restrict__ scores,
                      float* __restrict__ out_scores,   // B*K
                      int*   __restrict__ out_idx) {    // B*K (into d_out idx region)
  __shared__ float skey[N];
  __shared__ int   sidx[N];
  const int b = blockIdx.x;
  const float* s = scores + (size_t)b * N;

  for (int i = threadIdx.x; i < N; i += blockDim.x) {
    skey[i] = s[i];
    sidx[i] = i;
  }
  __syncthreads();

  // Bitonic sort, final order: descending by key, ties -> ascending index.
  for (int kk = 2; kk <= N; kk <<= 1) {
    for (int j = kk >> 1; j > 0; j >>= 1) {
      for (int t = threadIdx.x; t < N; t += blockDim.x) {
        const int l = t ^ j;
        if (l > t) {
          const float ka = skey[t], kb = skey[l];
          const int   ia = sidx[t], ib = sidx[l];
          const bool greater_t = (ka > kb) || (ka == kb && ia < ib);
          const bool desc      = ((t & kk) == 0);
          if (greater_t != desc) {
            skey[t] = kb; skey[l] = ka;
            sidx[t] = ib; sidx[l] = ia;
          }
        }
      }
      __syncthreads();
    }
  }

  for (int i = threadIdx.x; i < K; i += blockDim.x) {
    out_scores[(size_t)b * K + i] = skey[i];
    out_idx  [(size_t)b * K + i] = sidx[i];
  }
}

// ---------------------------------------------------------------------------
// Kernel 2: feature row gather. 256-thread blocks, 4 output rows per block.
// features is a 128 MB one-pass read and the output is write-once: both get
// non-temporal hints so they do not evict the adjacency working set from L2.
// ---------------------------------------------------------------------------
constexpr int FEAT_ROWS_PER_BLK = 4;

__global__ __launch_bounds__(256)
void feat_gather_kernel(const float* __restrict__ feats,
                        const int* __restrict__ idxs,
                        float* __restrict__ out) {
  const int lane = threadIdx.x & 63;              // 64 lanes per row (64*16B = 1KB)
  const int sub  = threadIdx.x >> 6;              // 0..3
  const int row  = blockIdx.x * FEAT_ROWS_PER_BLK + sub;   // 0 .. B*K-1
  const int b    = row >> 12;                     // row / K
  const int r    = idxs[row];
  const v4f* src = (const v4f*)(feats + ((size_t)b * N + (size_t)r) * C);
  v4f*       dst = (v4f*)(out + (size_t)row * C);
  v4f v = __builtin_nontemporal_load(&src[lane]);
  __builtin_nontemporal_store(v, &dst[lane]);
}

// ---------------------------------------------------------------------------
// Kernel 3: adjacency gather out[b,i,j] = adj[idx[b,i], idx[b,j]].
// Each WG handles 8 output rows of one batch:
//   - column-index table (16 KB) loaded once into LDS
//   - each 32 KB source row streamed into LDS with async B128 copies
//     (ASYNCcnt path), double-buffered: row rr+1 streams while rr is gathered
//   - random column gather served from LDS banks, output written coalesced
//     with non-temporal stores (1 GiB write-once stream must not evict the
//     256 MB adjacency matrix, whose rows are re-read ~8x across batches).
// LDS: 16 KB idx + 2 * 32 KB rows = 80 KB -> 4 WGs / WGP.
// ---------------------------------------------------------------------------
constexpr int ROWS_PER_WG = 8;
constexpr int ADJ_TPB     = 1024;

__device__ __forceinline__ void async_row_copy(const float* __restrict__ g,
                                               float* l, int tid) {
#if HAVE_ASYNC_LDS
  __attribute__((address_space(1))) char* gp =
      (__attribute__((address_space(1))) char*)g;          // strips const + casts AS
  __attribute__((address_space(3))) char* lp =
      (__attribute__((address_space(3))) char*)l;
  // 1024 lanes * 16 B = 16 KB per issue; two issues cover the 32 KB row.
  __builtin_amdgcn_global_load_async_to_lds_b128(
      (gv4i_t)(gp + (size_t)tid * 16), (lv4i_t)(lp + tid * 16), 0, 0);
  __builtin_amdgcn_global_load_async_to_lds_b128(
      (gv4i_t)(gp + 16384 + (size_t)tid * 16), (lv4i_t)(lp + 16384 + tid * 16), 0, 0);
#else
  v4f*       lv = (v4f*)l;
  const v4f* gv = (const v4f*)g;
  lv[tid]        = gv[tid];
  lv[tid + 1024] = gv[tid + 1024];
#endif
}

__global__ __launch_bounds__(ADJ_TPB)
void adj_gather_kernel(const float* __restrict__ adj,
                       const int* __restrict__ idxs,
                       float* __restrict__ out) {
  __shared__ int cidx[K];
  __shared__ alignas(16) float rowbuf[2][N];

  const int wg     = blockIdx.x;                 // 0 .. B*(K/ROWS_PER_WG)-1
  const int b      = wg / (K / ROWS_PER_WG);
  const int rowgrp = wg % (K / ROWS_PER_WG);
  const int i0     = rowgrp * ROWS_PER_WG;
  const int tid    = threadIdx.x;

  const int* bidx = idxs + (size_t)b * K;
  for (int j = tid; j < K; j += ADJ_TPB) cidx[j] = bidx[j];
  __syncthreads();

  // Prime the pipeline: row 0 into buffer 0.
  async_row_copy(adj + (size_t)cidx[i0] * N, rowbuf[0], tid);

  for (int rr = 0; rr < ROWS_PER_WG; ++rr) {
    if (rr + 1 < ROWS_PER_WG) {
      // Kick off next row into the other buffer, then wait for the *previous*
      // pair of async issues (in-order completion: cnt <= 2 means row rr done).
      async_row_copy(adj + (size_t)cidx[i0 + rr + 1] * N,
                     rowbuf[(rr + 1) & 1], tid);
      WAIT_ASYNC(2);
    } else {
      WAIT_ASYNC(0);
    }
    __syncthreads();   // all waves' async writes to rowbuf[rr&1] visible

    const float* rb = rowbuf[rr & 1];
    float* orow = out + ((size_t)b * K + (size_t)(i0 + rr)) * K;
    for (int j = tid; j < K; j += ADJ_TPB) {
      __builtin_nontemporal_store(rb[cidx[j]], &orow[j]);  // LDS gather -> NT store
    }
    __syncthreads();   // don't let anyone overwrite this buffer until all read
  }
}

// ---------------------------------------------------------------------------
// Launch
// ---------------------------------------------------------------------------
extern "C" void kernel_launch(void* const* d_in, const int* in_sizes, int n_in,
                              void* d_out, int out_size, void* d_ws, size_t ws_size,
                              hipStream_t stream) {
  const float* scores = (const float*)d_in[0];   // (B, N)
  const float* feats  = (const float*)d_in[1];   // (B, N, C)
  const float* adj    = (const float*)d_in[2];   // (N, N)
  float* out = (float*)d_out;

  float* out_feat   = out + OFF_FEAT;
  int*   out_idx    = (int*)(out + OFF_IDX);
  float* out_adj    = out + OFF_ADJ;
  float* out_scores = out + OFF_SC;

  // 1) per-batch sorted top-k (values + indices)
  topk_sort_kernel<<<B, 1024, 0, stream>>>(scores, out_scores, out_idx);

  // 2) feature row gather (reads indices produced on the same stream)
  feat_gather_kernel<<<(B * K) / FEAT_ROWS_PER_BLK, 256, 0, stream>>>(
      feats, (const int*)out_idx, out_feat);

  // 3) adjacency row+column gather with async LDS staging
  adj_gather_kernel<<<B * (K / ROWS_PER_WG), ADJ_TPB, 0, stream>>>(
      adj, (const int*)out_idx, out_adj);
}